// Pagerank_16492674417135
// MI455X (gfx1250) — compile-verified
//
#include <hip/hip_runtime.h>
#include <stdint.h>

// PageRank on MI455X (gfx1250). Memory-bound SpMV: WMMA inapplicable (random
// sparse indices, density 1.6e-5). CDNA5 paths: global_load_async_to_lds_b128
// bulk staging of the CSR column stream (+ s_wait_asynccnt), hardware global
// float/u32 atomics. CSR-by-destination built once (amortized over 100 iters);
// iterations are pure gather with the working set resident in the 192MB L2.

#define PR_ALPHA   0.85f
#define PR_ITERS   100
#define PR_BLK     256          // 8 waves (wave32)
#define NPB        1024         // nodes per block in k_iter; scan chunk size
#define CAP_WORDS  18432        // 72 KB LDS cols staging (mean 16384, ~16 sigma)

// ---------------- setup kernels ----------------

__global__ __launch_bounds__(PR_BLK) void k_prep(uint32_t* __restrict__ odeg,
                                                 uint32_t* __restrict__ rdeg,
                                                 int* done, float* err, uint32_t* ticket,
                                                 int N, int useCsr) {
  int stride = gridDim.x * blockDim.x;
  for (int i = blockIdx.x * blockDim.x + threadIdx.x; i < N; i += stride) {
    odeg[i] = 0u;
    if (useCsr) rdeg[i] = 0u;
  }
  if (blockIdx.x == 0 && threadIdx.x == 0) { *done = 0; *err = 0.0f; *ticket = 0u; }
}

// One pass over int64 edges: in-degree of dst rows, out-degree of src cols.
__global__ __launch_bounds__(PR_BLK) void k_count(const long long* __restrict__ ei,
                                                  uint32_t* __restrict__ odeg,
                                                  uint32_t* __restrict__ rdeg,
                                                  long long E) {
  long long stride = (long long)gridDim.x * blockDim.x;
  for (long long e = blockIdx.x * (long long)blockDim.x + threadIdx.x; e < E; e += stride) {
    uint32_t r = (uint32_t)ei[e];
    uint32_t c = (uint32_t)ei[E + e];
    __hip_atomic_fetch_add(&rdeg[r], 1u, __ATOMIC_RELAXED, __HIP_MEMORY_SCOPE_AGENT);
    __hip_atomic_fetch_add(&odeg[c], 1u, __ATOMIC_RELAXED, __HIP_MEMORY_SCOPE_AGENT);
  }
}

__global__ __launch_bounds__(PR_BLK) void k_deg_only(const long long* __restrict__ ei,
                                                     uint32_t* __restrict__ odeg,
                                                     long long E) {
  long long stride = (long long)gridDim.x * blockDim.x;
  for (long long e = blockIdx.x * (long long)blockDim.x + threadIdx.x; e < E; e += stride) {
    uint32_t c = (uint32_t)ei[E + e];
    __hip_atomic_fetch_add(&odeg[c], 1u, __ATOMIC_RELAXED, __HIP_MEMORY_SCOPE_AGENT);
  }
}

// chunk sums of rdeg (chunk = NPB)
__global__ __launch_bounds__(PR_BLK) void k_scan1(const uint32_t* __restrict__ rdeg,
                                                  uint32_t* __restrict__ csum, int N) {
  int base = blockIdx.x * NPB;
  uint32_t s = 0u;
  for (int i = threadIdx.x; i < NPB; i += PR_BLK) {
    int g = base + i;
    if (g < N) s += rdeg[g];
  }
  __shared__ uint32_t red[PR_BLK];
  red[threadIdx.x] = s; __syncthreads();
  for (int st = PR_BLK / 2; st > 0; st >>= 1) {
    if (threadIdx.x < st) red[threadIdx.x] += red[threadIdx.x + st];
    __syncthreads();
  }
  if (threadIdx.x == 0) csum[blockIdx.x] = red[0];
}

// single-block exclusive scan of chunk sums (C <= 1024; 4 per thread)
__global__ __launch_bounds__(PR_BLK) void k_scan2(const uint32_t* __restrict__ csum,
                                                  uint32_t* __restrict__ coff, int C) {
  uint32_t vals[4];
  int b0 = threadIdx.x * 4;
  for (int j = 0; j < 4; ++j) vals[j] = (b0 + j < C) ? csum[b0 + j] : 0u;
  uint32_t tsum = vals[0] + vals[1] + vals[2] + vals[3];
  __shared__ uint32_t ss[PR_BLK];
  ss[threadIdx.x] = tsum; __syncthreads();
  for (int d = 1; d < PR_BLK; d <<= 1) {
    uint32_t t = (threadIdx.x >= d) ? ss[threadIdx.x - d] : 0u;
    __syncthreads();
    ss[threadIdx.x] += t;
    __syncthreads();
  }
  uint32_t run = ss[threadIdx.x] - tsum;   // exclusive
  for (int j = 0; j < 4; ++j) {
    if (b0 + j < C) coff[b0 + j] = run;
    run += vals[j];
  }
}

// per-chunk exclusive scan -> rowptr (and cursor copy); rowptr[N] = E
__global__ __launch_bounds__(PR_BLK) void k_scan3(const uint32_t* __restrict__ rdeg,
                                                  const uint32_t* __restrict__ coff,
                                                  uint32_t* __restrict__ rowptr,
                                                  uint32_t* __restrict__ cursor, int N) {
  int base = blockIdx.x * NPB;
  int b0 = base + threadIdx.x * 4;
  uint32_t vals[4];
  for (int j = 0; j < 4; ++j) vals[j] = (b0 + j < N) ? rdeg[b0 + j] : 0u;
  uint32_t tsum = vals[0] + vals[1] + vals[2] + vals[3];
  __shared__ uint32_t ss[PR_BLK];
  ss[threadIdx.x] = tsum; __syncthreads();
  for (int d = 1; d < PR_BLK; d <<= 1) {
    uint32_t t = (threadIdx.x >= d) ? ss[threadIdx.x - d] : 0u;
    __syncthreads();
    ss[threadIdx.x] += t;
    __syncthreads();
  }
  uint32_t run = (ss[threadIdx.x] - tsum) + coff[blockIdx.x];
  for (int j = 0; j < 4; ++j) {
    int i = b0 + j;
    if (i < N) {
      rowptr[i] = run;
      cursor[i] = run;
      if (i == N - 1) rowptr[N] = run + vals[j];
    }
    run += vals[j];
  }
}

// scatter cols into CSR slots (16M u32 atomics, once per call)
__global__ __launch_bounds__(PR_BLK) void k_fill(const long long* __restrict__ ei,
                                                 uint32_t* __restrict__ cursor,
                                                 uint32_t* __restrict__ cols, long long E) {
  long long stride = (long long)gridDim.x * blockDim.x;
  for (long long e = blockIdx.x * (long long)blockDim.x + threadIdx.x; e < E; e += stride) {
    uint32_t r = (uint32_t)ei[e];
    uint32_t c = (uint32_t)ei[E + e];
    uint32_t slot = __hip_atomic_fetch_add(&cursor[r], 1u,
                                           __ATOMIC_RELAXED, __HIP_MEMORY_SCOPE_AGENT);
    cols[slot] = c;
  }
}

// invd = ALPHA/deg (bit-identical to reference vals); v = 1/N; w = v*invd
__global__ __launch_bounds__(PR_BLK) void k_nodeinit(const uint32_t* __restrict__ odeg,
                                                     float* __restrict__ invd,
                                                     float* __restrict__ w,
                                                     float* __restrict__ v,
                                                     int N, float inv_n) {
  int stride = gridDim.x * blockDim.x;
  for (int i = blockIdx.x * blockDim.x + threadIdx.x; i < N; i += stride) {
    float iv = PR_ALPHA / (float)odeg[i];   // deg==0 -> inf, never gathered
    invd[i] = iv;
    v[i] = inv_n;
    w[i] = inv_n * iv;
  }
}

// 4-way ILP gather: independent accumulators keep 4 L2 gathers in flight.
__device__ __forceinline__ float gather_sum(const uint32_t* __restrict__ idx,
                                            uint32_t e, uint32_t eend,
                                            const float* __restrict__ wOld) {
  float a0 = 0.0f, a1 = 0.0f, a2 = 0.0f, a3 = 0.0f;
  for (; e + 4 <= eend; e += 4) {
    uint32_t c0 = idx[e], c1 = idx[e + 1], c2 = idx[e + 2], c3 = idx[e + 3];
    a0 += wOld[c0]; a1 += wOld[c1]; a2 += wOld[c2]; a3 += wOld[c3];
  }
  for (; e < eend; ++e) a0 += wOld[idx[e]];
  return (a0 + a1) + (a2 + a3);
}

// ---------------- main iteration kernel (CSR gather) ----------------
// One launch per PageRank step. Block owns NPB consecutive nodes and async-
// stages its contiguous cols range into LDS. Fused error reduction and
// convergence check via last-block ticket.
__global__ __launch_bounds__(PR_BLK) void k_iter(const uint32_t* __restrict__ rowptr,
                                                 const uint32_t* __restrict__ cols,
                                                 const float* __restrict__ wOld,
                                                 float* __restrict__ wNew,
                                                 float* __restrict__ v,
                                                 const float* __restrict__ invd,
                                                 float* err, uint32_t* ticket, int* done,
                                                 int N, float teleport, float thresh) {
  if (__hip_atomic_load(done, __ATOMIC_RELAXED, __HIP_MEMORY_SCOPE_AGENT)) return;

  __shared__ alignas(16) uint32_t sCols[CAP_WORDS];   // 72 KB
  __shared__ uint32_t sRow[NPB + 1];
  __shared__ float ssum[PR_BLK];

  const int tid  = threadIdx.x;
  const int base = blockIdx.x * NPB;
  const int cnt  = (N - base < NPB) ? (N - base) : NPB;

  const uint32_t start = rowptr[base];
  const uint32_t end   = rowptr[base + cnt];
  const uint32_t w0    = start & ~3u;            // 16B-align the staged range
  const uint32_t w1    = (end + 3u) & ~3u;
  const uint32_t nbytes = (w1 - w0) * 4u;        // multiple of 16
  const bool staged = (w1 - w0) <= (uint32_t)CAP_WORDS;

  if (staged) {
    // per-lane 16B async copies: wave moves 512B/inst, ideal coalescing
    uint64_t g = (uint64_t)(uintptr_t)cols + (uint64_t)w0 * 4u;
    uint32_t l = (uint32_t)(uintptr_t)&sCols[0];   // low 32 bits = LDS offset
    for (uint32_t ofs = (uint32_t)tid * 16u; ofs < nbytes; ofs += PR_BLK * 16u) {
      uint32_t lj = l + ofs;
      uint64_t gj = g + ofs;
      asm volatile("global_load_async_to_lds_b128 %0, %1, off"
                   :: "v"(lj), "v"(gj) : "memory");
    }
  }
  // overlap: fill rowptr tile with normal loads while async engine streams cols
  for (int i = tid; i <= cnt; i += PR_BLK) sRow[i] = rowptr[base + i];
  if (staged) asm volatile("s_wait_asynccnt 0" ::: "memory");
  __syncthreads();

  float partial = 0.0f;
#pragma unroll
  for (int k = 0; k < NPB / PR_BLK; ++k) {
    int li = k * PR_BLK + tid;
    if (li < cnt) {
      int n = base + li;
      uint32_t s0 = sRow[li], e0 = sRow[li + 1];
      float s = staged ? gather_sum(sCols, s0 - w0, e0 - w0, wOld)
                       : gather_sum(cols, s0, e0, wOld);
      float nv = s + teleport;
      partial += fabsf(nv - v[n]);
      v[n]    = nv;              // safe: only this thread touches node n
      wNew[n] = nv * invd[n];    // ping-pong buffer, no cross-thread hazard
    }
  }

  ssum[tid] = partial; __syncthreads();
  for (int st = PR_BLK / 2; st > 0; st >>= 1) {
    if (tid < st) ssum[tid] += ssum[tid + st];
    __syncthreads();
  }
  if (tid == 0) {
    __hip_atomic_fetch_add(err, ssum[0], __ATOMIC_RELAXED, __HIP_MEMORY_SCOPE_AGENT);
    uint32_t t = __hip_atomic_fetch_add(ticket, 1u, __ATOMIC_ACQ_REL, __HIP_MEMORY_SCOPE_AGENT);
    if (t == (uint32_t)gridDim.x - 1u) {    // last block: check + reset
      float etot = __hip_atomic_load(err, __ATOMIC_RELAXED, __HIP_MEMORY_SCOPE_AGENT);
      if (etot < thresh)
        __hip_atomic_store(done, 1, __ATOMIC_RELAXED, __HIP_MEMORY_SCOPE_AGENT);
      __hip_atomic_store(err, 0.0f, __ATOMIC_RELAXED, __HIP_MEMORY_SCOPE_AGENT);
      __hip_atomic_store(ticket, 0u, __ATOMIC_RELAXED, __HIP_MEMORY_SCOPE_AGENT);
    }
  }
}

// ---------------- fallback path (workspace too small for CSR) ----------------

__global__ __launch_bounds__(PR_BLK) void k_zero(float* __restrict__ acc, float* err,
                                                 const int* __restrict__ done, int N) {
  if (__hip_atomic_load(done, __ATOMIC_RELAXED, __HIP_MEMORY_SCOPE_AGENT)) return;
  int stride = gridDim.x * blockDim.x;
  for (int i = blockIdx.x * blockDim.x + threadIdx.x; i < N; i += stride) acc[i] = 0.0f;
  if (blockIdx.x == 0 && threadIdx.x == 0) *err = 0.0f;
}

__global__ __launch_bounds__(PR_BLK) void k_scatter_i64(const long long* __restrict__ ei,
                                                        const float* __restrict__ w,
                                                        float* __restrict__ acc,
                                                        const int* __restrict__ done,
                                                        long long E) {
  if (__hip_atomic_load(done, __ATOMIC_RELAXED, __HIP_MEMORY_SCOPE_AGENT)) return;
  long long stride = (long long)gridDim.x * blockDim.x;
  for (long long e = blockIdx.x * (long long)blockDim.x + threadIdx.x; e < E; e += stride) {
    uint32_t r = (uint32_t)ei[e];
    uint32_t c = (uint32_t)ei[E + e];
    __hip_atomic_fetch_add(&acc[r], w[c], __ATOMIC_RELAXED, __HIP_MEMORY_SCOPE_AGENT);
  }
}

__global__ __launch_bounds__(PR_BLK) void k_commit(const float* __restrict__ acc,
                                                   float* __restrict__ v,
                                                   float* __restrict__ w,
                                                   const float* __restrict__ invd,
                                                   float* err, const int* __restrict__ done,
                                                   int N, float teleport) {
  if (__hip_atomic_load(done, __ATOMIC_RELAXED, __HIP_MEMORY_SCOPE_AGENT)) return;
  float partial = 0.0f;
  int stride = gridDim.x * blockDim.x;
  for (int i = blockIdx.x * blockDim.x + threadIdx.x; i < N; i += stride) {
    float nv = acc[i] + teleport;
    partial += fabsf(nv - v[i]);
    v[i] = nv;
    w[i] = nv * invd[i];
  }
  __shared__ float red[PR_BLK];
  red[threadIdx.x] = partial; __syncthreads();
  for (int s = PR_BLK / 2; s > 0; s >>= 1) {
    if (threadIdx.x < s) red[threadIdx.x] += red[threadIdx.x + s];
    __syncthreads();
  }
  if (threadIdx.x == 0)
    __hip_atomic_fetch_add(err, red[0], __ATOMIC_RELAXED, __HIP_MEMORY_SCOPE_AGENT);
}

__global__ void k_check(const float* __restrict__ err, int* done, float thresh) {
  if (threadIdx.x == 0 && blockIdx.x == 0) {
    if (*err < thresh) *done = 1;
  }
}

// ---------------- host launcher ----------------

extern "C" void kernel_launch(void* const* d_in, const int* in_sizes, int n_in,
                              void* d_out, int out_size, void* d_ws, size_t ws_size,
                              hipStream_t stream) {
  (void)n_in; (void)out_size;
  const long long* ei = (const long long*)d_in[1];   // int64 edge_index [2,E]
  const int N = in_sizes[0];
  const long long E = (long long)in_sizes[1] / 2;
  float* v = (float*)d_out;

  const size_t N4 = (size_t)N * 4;
  char* ws = (char*)d_ws;
  size_t off = 0;
  uint32_t* odeg   = (uint32_t*)(ws + off); off += N4;
  float*    invd   = (float*)   (ws + off); off += N4;
  float*    wA     = (float*)   (ws + off); off += N4;
  float*    wB     = (float*)   (ws + off); off += N4;   // acc in fallback
  float*    err    = (float*)   (ws + off); off += 64;
  int*      done   = (int*)     (ws + off); off += 64;
  uint32_t* ticket = (uint32_t*)(ws + off); off += 64;
  // CSR-only arrays
  uint32_t* rdeg   = (uint32_t*)(ws + off); off += N4;
  uint32_t* rowptr = (uint32_t*)(ws + off); off += N4 + 64;   // N+1 entries
  uint32_t* cursor = (uint32_t*)(ws + off); off += N4;
  uint32_t* csum   = (uint32_t*)(ws + off); off += 4096;
  uint32_t* coff   = (uint32_t*)(ws + off); off += 4096;
  size_t colsOff = (off + 63) & ~(size_t)63;                  // 16B+ aligned
  uint32_t* cols   = (uint32_t*)(ws + colsOff);

  const int nChunks = (N + NPB - 1) / NPB;                    // <= 1024 for N <= 1M
  const bool useCsr = (nChunks <= 1024) &&
                      (ws_size >= colsOff + (size_t)E * 4 + 16);

  const float inv_n    = 1.0f / (float)N;
  const float teleport = (1.0f - PR_ALPHA) * inv_n;
  const float thresh   = (float)N * 1e-6f;

  int nb = (N + PR_BLK - 1) / PR_BLK; if (nb > 4096) nb = 4096;

  k_prep<<<nb, PR_BLK, 0, stream>>>(odeg, rdeg, done, err, ticket, N, useCsr ? 1 : 0);

  if (useCsr) {
    k_count<<<4096, PR_BLK, 0, stream>>>(ei, odeg, rdeg, E);
    k_scan1<<<nChunks, PR_BLK, 0, stream>>>(rdeg, csum, N);
    k_scan2<<<1, PR_BLK, 0, stream>>>(csum, coff, nChunks);
    k_scan3<<<nChunks, PR_BLK, 0, stream>>>(rdeg, coff, rowptr, cursor, N);
    k_fill<<<4096, PR_BLK, 0, stream>>>(ei, cursor, cols, E);
    k_nodeinit<<<nb, PR_BLK, 0, stream>>>(odeg, invd, wA, v, N, inv_n);
    for (int it = 0; it < PR_ITERS; ++it) {
      const float* wo = (it & 1) ? wB : wA;
      float*       wn = (it & 1) ? wA : wB;
      k_iter<<<nChunks, PR_BLK, 0, stream>>>(rowptr, cols, wo, wn, v, invd,
                                             err, ticket, done, N, teleport, thresh);
    }
  } else {
    k_deg_only<<<4096, PR_BLK, 0, stream>>>(ei, odeg, E);
    k_nodeinit<<<nb, PR_BLK, 0, stream>>>(odeg, invd, wA, v, N, inv_n);
    for (int it = 0; it < PR_ITERS; ++it) {
      k_zero<<<nb, PR_BLK, 0, stream>>>(wB, err, done, N);
      k_scatter_i64<<<4096, PR_BLK, 0, stream>>>(ei, wA, wB, done, E);
      k_commit<<<1024, PR_BLK, 0, stream>>>(wB, v, wA, invd, err, done, N, teleport);
      k_check<<<1, 32, 0, stream>>>(err, done, thresh);
    }
  }
}